// TensorRingLinear_50955491999867
// MI455X (gfx1250) — compile-verified
//
#include <hip/hip_runtime.h>

// TensorRingLinear fused kernel for MI455X (gfx1250, wave32, WMMA).
// One workgroup per batch element; whole 8-node tensor-ring chain kept in LDS
// (~112 KB of the 320 KB/WGP), all GEMMs via V_WMMA_F32_16X16X4_F32 (exact
// fp32 semantics). node0+node1 and node6+node7 are fused so the two 256KB
// intermediates are never materialized (~512MB of HBM traffic avoided).
// K-loops are chunked by 16 with register preload of 4 A/B fragments so the
// compiler can batch LDS loads ahead of one s_wait_dscnt and issue 4 WMMAs
// back-to-back (latency hiding + XDL co-execution).

typedef float v2f __attribute__((ext_vector_type(2)));
typedef float v4f __attribute__((ext_vector_type(4)));
typedef float v8f __attribute__((ext_vector_type(8)));

#define NW 8  // waves per block (256 threads / wave32)

__device__ __forceinline__ v8f wmma4(v2f a, v2f b, v8f c) {
  // D(16x16,f32) = A(16x4,f32) * B(4x16,f32) + C
  return __builtin_amdgcn_wmma_f32_16x16x4_f32(
      false, a, false, b, (short)0, c, false, false);
}

// B fragment (4x16): B[k][n] = W[(n0+n)*Kw + k0+k], W row-major (N x K) global.
__device__ __forceinline__ v2f ldB_w(const float* __restrict__ W, int Kw,
                                     int n0, int k0, int lane) {
  int n = lane & 15;
  int kk = (lane >> 4) << 1;
  const float* p = W + (n0 + n) * Kw + (k0 + kk);
  v2f r; r[0] = p[0]; r[1] = p[1];
  return r;
}

// A fragment (16x4) from LDS matrix, leading dim ld: A[m0+r][k0+k].
__device__ __forceinline__ v2f ldA(const float* A, int ld, int m0, int k0, int lane) {
  int r = lane & 15;
  int kk = (lane >> 4) << 1;
  const float* p = A + (m0 + r) * ld + (k0 + kk);
  v2f v; v[0] = p[0]; v[1] = p[1];
  return v;
}

// Store 16x16 f32 D tile to LDS: lane L, vgpr i -> row m0+i+8*(L>>4), col n0+(L&15).
__device__ __forceinline__ void stD(float* D, int ld, int m0, int n0, int lane, v8f acc) {
  int n = n0 + (lane & 15);
  int r0 = m0 + ((lane >> 4) << 3);
#pragma unroll
  for (int i = 0; i < 8; ++i) D[(r0 + i) * ld + n] = acc[i];
}

// Y (MxN, ld=N, LDS) = A (MxK, ld=K, LDS) @ W.T ; W is (N x K) row-major global.
// K must be a multiple of 16: chunked 4-frag preload per 16 K-values.
__device__ __forceinline__ void gemm(const float* A, float* Y,
                                     const float* __restrict__ W,
                                     int M, int N, int K, int lane, int wid) {
  int mt = M >> 4, nt = N >> 4;
  for (int t = wid; t < mt * nt; t += NW) {
    int m0 = (t / nt) << 4, n0 = (t % nt) << 4;
    v8f acc = {0.f, 0.f, 0.f, 0.f, 0.f, 0.f, 0.f, 0.f};
    for (int k0 = 0; k0 < K; k0 += 16) {
      v2f a[4], bf[4];
#pragma unroll
      for (int i = 0; i < 4; ++i) {
        a[i]  = ldA(A, K, m0, k0 + 4 * i, lane);
        bf[i] = ldB_w(W, K, n0, k0 + 4 * i, lane);
      }
#pragma unroll
      for (int i = 0; i < 4; ++i) acc = wmma4(a[i], bf[i], acc);
    }
    stD(Y, N, m0, n0, lane, acc);
  }
}

__global__ __launch_bounds__(256)
void TensorRingLinear_50955491999867_kernel(
    const float* __restrict__ x,
    const float* __restrict__ W0, const float* __restrict__ W1,
    const float* __restrict__ W2, const float* __restrict__ W3,
    const float* __restrict__ W4, const float* __restrict__ W5,
    const float* __restrict__ W6, const float* __restrict__ W7,
    const float* __restrict__ bias, float* __restrict__ out) {
  extern __shared__ float smem[];
  float* M0  = smem;          // 1024  : permuted input (256 x 4)
  float* Y2  = smem + 1024;   // 1024  : (64 x 16)
  float* Y3  = smem + 2048;   // 256   : (16 x 16)
  float* Y4  = smem + 2560;   // 1024  : (16 x 64)
  float* BIG = smem + 3584;   // 8192  : Y1 (512x16), later Y5 (64x128)
  float* T   = smem + 11776;  // 16384 : front T[4][256][16]; back Y6blk(4096)+partials(2048)

  const int b    = blockIdx.x;
  const int tid  = threadIdx.x;
  const int lane = tid & 31;
  const int wid  = tid >> 5;

  // ---- load x_b and apply the mode-reversal permute into M0 (256 x 4) ----
  // x flat idx = ((m0*8+m1)*8+m2)*4+m3 ; M0 row r = m3*64+m2*8+m1, col m0.
  {
    v4f xv = *(const v4f*)(x + b * 1024 + tid * 4);  // coalesced 16B/lane
    int m0v = tid >> 6, m1 = (tid >> 3) & 7, m2 = tid & 7;
#pragma unroll
    for (int m3 = 0; m3 < 4; ++m3)
      M0[(m3 * 64 + m2 * 8 + m1) * 4 + m0v] = xv[m3];
  }
  __syncthreads();

  // ---- fused node0+node1 -> Y1 (512 x 16) in BIG ----
  // For c0 in 0..15: T_c0(256x16) = M0 @ W0^T[:, {j*16+c0}]; then
  // Y1 rows [c0*32, +32) = (T_c0 viewed flat as 32x128) @ W1^T(128x16).
  for (int cg = 0; cg < 4; ++cg) {
    // step A: 4 c0's x 16 M-tiles = 64 single-K-step WMMA tiles, 8 per wave
    for (int t = wid; t < 64; t += NW) {
      int cl  = t >> 4;           // local c0 index 0..3
      int m0t = (t & 15) << 4;
      int c0  = cg * 4 + cl;
      v2f a = ldA(M0, 4, m0t, 0, lane);
      // B[k][j] = W0[(j*16+c0)*4 + k]   (W0 is (256 x 4) row-major)
      int j = lane & 15, kk = (lane >> 4) << 1;
      const float* p = W0 + (j * 16 + c0) * 4 + kk;
      v2f bf; bf[0] = p[0]; bf[1] = p[1];
      v8f acc = {0.f, 0.f, 0.f, 0.f, 0.f, 0.f, 0.f, 0.f};
      acc = wmma4(a, bf, acc);
      stD(T + cl * 4096, 16, m0t, 0, lane, acc);
    }
    __syncthreads();
    // step B: 4 c0's x 2 M-tiles = 8 tiles, 1 per wave, K = 128 (chunked by 16)
    {
      int cl  = wid >> 1;
      int m0t = (wid & 1) << 4;
      int c0  = cg * 4 + cl;
      const float* Acl = T + cl * 4096;  // flat view: (32 x 128) row-major
      v8f acc = {0.f, 0.f, 0.f, 0.f, 0.f, 0.f, 0.f, 0.f};
      for (int k0 = 0; k0 < 128; k0 += 16) {
        v2f a[4], bf[4];
#pragma unroll
        for (int i = 0; i < 4; ++i) {
          a[i]  = ldA(Acl, 128, m0t, k0 + 4 * i, lane);
          bf[i] = ldB_w(W1, 128, 0, k0 + 4 * i, lane);
        }
#pragma unroll
        for (int i = 0; i < 4; ++i) acc = wmma4(a[i], bf[i], acc);
      }
      stD(BIG + c0 * 32 * 16, 16, m0t, 0, lane, acc);
    }
    __syncthreads();
  }

  // ---- middle nodes: plain flat-reshape GEMMs, intermediates in LDS ----
  gemm(BIG, Y2, W2, 64, 16, 128, lane, wid);   // node2: (64x128)@(128x16)
  __syncthreads();
  gemm(Y2, Y3, W3, 16, 16, 64, lane, wid);     // node3: (16x64)@(64x16)
  __syncthreads();
  gemm(Y3, Y4, W4, 16, 64, 16, lane, wid);     // node4: (16x16)@(16x64)
  __syncthreads();
  gemm(Y4, BIG, W5, 64, 128, 16, lane, wid);   // node5: (64x16)@(16x128) -> Y5 overwrites BIG
  __syncthreads();

  // ---- fused node6+node7 -> out rows, 16 output rows per gp iteration ----
  float* Y5    = BIG;       // flat view (512 x 16)
  float* Y6blk = T;         // (32 x 128)
  float* part  = T + 4096;  // 8 partial (16x16) tiles for split-K reduce

  // gp-invariant weight fragments, held in VGPRs across the whole back loop:
  // W7 B-frags for this wave's split-K chunk (N padded 4 -> 16, zeros beyond).
  v2f b7[8];
  {
    int n = lane & 15;
#pragma unroll
    for (int i = 0; i < 8; ++i) {
      b7[i][0] = 0.f; b7[i][1] = 0.f;
      if (n < 4) {
        int kk = wid * 32 + i * 4 + ((lane >> 4) << 1);
        const float* q = W7 + n * 256 + kk;
        b7[i][0] = q[0]; b7[i][1] = q[1];
      }
    }
  }
  // W6 B-frags: both of this wave's stepA tiles share n0t = wid*16.
  v2f b6[4];
#pragma unroll
  for (int i = 0; i < 4; ++i) b6[i] = ldB_w(W6, 16, wid << 4, i * 4, lane);

  for (int gp = 0; gp < 16; ++gp) {
    // step A: Y6blk(32x128) = A6pair(32x16) @ W6^T(16x128); this wave owns
    // tiles (m0t=0, n0t=wid*16) and (m0t=16, n0t=wid*16).
    // A6pair[row][k] = Y5flat[ ((row>>1)*32 + 2*gp + (row&1))*16 + k ]
#pragma unroll
    for (int tt = 0; tt < 2; ++tt) {
      int m0t = tt << 4;
      int row = m0t + (lane & 15);
      int rY5 = ((row >> 1) * 32) + 2 * gp + (row & 1);
      v2f a[4];
#pragma unroll
      for (int ks = 0; ks < 4; ++ks) {
        const float* p = Y5 + rY5 * 16 + ks * 4 + ((lane >> 4) << 1);
        a[ks][0] = p[0]; a[ks][1] = p[1];
      }
      v8f acc = {0.f, 0.f, 0.f, 0.f, 0.f, 0.f, 0.f, 0.f};
#pragma unroll
      for (int ks = 0; ks < 4; ++ks) acc = wmma4(a[ks], b6[ks], acc);
      stD(Y6blk, 128, m0t, wid << 4, lane, acc);
    }
    __syncthreads();
    // step B: split-K over 8 waves: each does a K-chunk of 32 of
    // (16x256) A7tile @ W7^T(padded 256x16); partials reduced in LDS.
    {
      v8f acc = {0.f, 0.f, 0.f, 0.f, 0.f, 0.f, 0.f, 0.f};
      int ur = lane & 15;
#pragma unroll
      for (int g2 = 0; g2 < 2; ++g2) {
        v2f af[4];
#pragma unroll
        for (int i = 0; i < 4; ++i) {
          int kk = wid * 32 + g2 * 16 + i * 4 + ((lane >> 4) << 1);
          int a_ = kk >> 4, c_ = kk & 15;
          const float* p = Y6blk + ((a_ << 1) + (ur >> 3)) * 128 + ((ur & 7) << 4) + c_;
          af[i][0] = p[0]; af[i][1] = p[1];
        }
#pragma unroll
        for (int i = 0; i < 4; ++i) acc = wmma4(af[i], b7[g2 * 4 + i], acc);
      }
      stD(part + wid * 256, 16, 0, 0, lane, acc);
    }
    __syncthreads();
    // reduce 8 partials + bias; out rows u = 16*gp..+15 are 64 contiguous
    // floats -> threads 0..63 store one float each (coalesced 256B burst).
    if (tid < 64) {
      int ur = tid >> 2, n = tid & 3;
      float s = 0.f;
#pragma unroll
      for (int w = 0; w < 8; ++w) s += part[w * 256 + ur * 16 + n];
      out[b * 1024 + gp * 64 + tid] = s + bias[gp * 64 + tid];
    }
    __syncthreads();
  }
}

extern "C" void kernel_launch(void* const* d_in, const int* in_sizes, int n_in,
                              void* d_out, int out_size, void* d_ws, size_t ws_size,
                              hipStream_t stream) {
  (void)in_sizes; (void)n_in; (void)out_size; (void)d_ws; (void)ws_size;
  const float* x  = (const float*)d_in[0];
  const float* W0 = (const float*)d_in[1];
  const float* W1 = (const float*)d_in[2];
  const float* W2 = (const float*)d_in[3];
  const float* W3 = (const float*)d_in[4];
  const float* W4 = (const float*)d_in[5];
  const float* W5 = (const float*)d_in[6];
  const float* W6 = (const float*)d_in[7];
  const float* W7 = (const float*)d_in[8];
  const float* bias = (const float*)d_in[9];
  float* out = (float*)d_out;

  const size_t shmem = 28160 * sizeof(float);  // 112,640 B of the 320 KB/WGP LDS
  (void)hipFuncSetAttribute(
      reinterpret_cast<const void*>(TensorRingLinear_50955491999867_kernel),
      hipFuncAttributeMaxDynamicSharedMemorySize, (int)shmem);

  TensorRingLinear_50955491999867_kernel<<<1024, 256, shmem, stream>>>(
      x, W0, W1, W2, W3, W4, W5, W6, W7, bias, out);
}